// FeatureReplacementWrapper_63857573757012
// MI455X (gfx1250) — compile-verified
//
#include <hip/hip_runtime.h>

namespace {
constexpr int kC          = 512;            // channels
constexpr int kHW4        = (28 * 28) / 4;  // 196 float4 per (b,c) plane
constexpr int kNumReplace = 256;
constexpr int kBlock      = 256;            // 8 wave32s per block
}

typedef __attribute__((ext_vector_type(4))) int v4i;
typedef __attribute__((address_space(1))) v4i gv4i;   // global int4
typedef __attribute__((address_space(3))) v4i sv4i;   // LDS int4

// Kernel 1: build a 512-byte per-channel replacement mask in workspace.
__global__ __launch_bounds__(512) void fr_build_mask(
    const int* __restrict__ idx, unsigned char* __restrict__ mask) {
  const int t = threadIdx.x;
  if (t < kC) mask[t] = 0;
  __syncthreads();
  if (t < kNumReplace) mask[idx[t] & (kC - 1)] = 1;  // benign races: all write 1
}

// Kernel 2: one float4 per thread; read exactly one source per element.
__global__ __launch_bounds__(kBlock) void fr_select(
    const float4* __restrict__ base, const float4* __restrict__ feat,
    const unsigned char* __restrict__ mask, float4* __restrict__ dst,
    unsigned int n4) {
  const unsigned int i = blockIdx.x * (unsigned)kBlock + threadIdx.x;
  if (i >= n4) return;

  const unsigned int plane = i / (unsigned)kHW4;       // b*C + c (mul-hi, no div)
  const unsigned int c     = plane & (unsigned)(kC - 1);
  const float4* __restrict__ src = mask[c] ? feat : base;  // branchless ptr select

#if defined(__gfx1250__) && __has_builtin(__builtin_amdgcn_global_load_async_to_lds_b128)
  // gfx1250 async data-mover path: global -> LDS (ASYNCcnt), then LDS -> regs -> store.
  __shared__ float4 stage[kBlock];
  __builtin_amdgcn_global_load_async_to_lds_b128(
      (gv4i*)(src + i),
      (sv4i*)&stage[threadIdx.x],
      /*offset=*/0, /*cpol=*/0);
#if __has_builtin(__builtin_amdgcn_s_wait_asynccnt)
  __builtin_amdgcn_s_wait_asynccnt(0);
#else
  asm volatile("s_wait_asynccnt 0" ::: "memory");
#endif
  dst[i] = stage[threadIdx.x];  // ds_load_b128 + global_store_b128
#else
  dst[i] = src[i];              // global_load_b128 + global_store_b128
#endif
}

extern "C" void kernel_launch(void* const* d_in, const int* in_sizes, int n_in,
                              void* d_out, int out_size, void* d_ws, size_t ws_size,
                              hipStream_t stream) {
  const float* base = (const float*)d_in[0];          // output [B,C,H,W] fp32
  const float* feat = (const float*)d_in[1];          // matryoshka_features fp32
  const int*   idx  = (const int*)d_in[2];            // indices_to_replace
  unsigned char* mask = (unsigned char*)d_ws;         // 512 bytes of scratch
  float* out = (float*)d_out;

  fr_build_mask<<<1, 512, 0, stream>>>(idx, mask);

  const unsigned int n4   = (unsigned int)out_size / 4u;   // 3,211,264 float4s
  const unsigned int grid = (n4 + (unsigned)kBlock - 1) / (unsigned)kBlock;
  fr_select<<<grid, kBlock, 0, stream>>>(
      (const float4*)base, (const float4*)feat, mask, (float4*)out, n4);
}